// TropicalLinear_23295902613810
// MI455X (gfx1250) — compile-verified
//
#include <hip/hip_runtime.h>
#include <hip/hip_bf16.h>
#include <math.h>

// Tropical (max-plus) linear layer, forward value:
//   out[n,o] = max_i( w[o,i] + x[n,i] ) + bias[o]
// (the softmax STE term cancels exactly in the forward value).
//
// CDNA5: async global->LDS staging (GLOBAL_LOAD_ASYNC_TO_LDS_B128 /
// ASYNCcnt), double-buffered to overlap copy of chunk k+1 with compute of
// chunk k. 2x2 register blocking per thread (32x32 output tile per block)
// halves LDS traffic per FLOP; fmax(fmax(a,b),acc) pattern fuses to
// v_max3_f32. WMMA is not applicable to the (max,+) semiring.

#define TN 32          // n-tile (rows of x per block)
#define TO 32          // o-tile (rows of w per block)
#define KC 64          // K chunk (floats)
#define PAD 4          // +4 floats -> 4-bank shift per row, keeps b128 alignment
#define KCP (KC + PAD)

#if defined(__gfx1250__) && __has_builtin(__builtin_amdgcn_global_load_async_to_lds_b128)
#define HAVE_ASYNC_LDS 1
#endif

typedef int v4i __attribute__((ext_vector_type(4)));

__device__ __forceinline__ void async_copy16(const float* g, float* l) {
#ifdef HAVE_ASYNC_LDS
    // Signature confirmed by clang: (v4i AS1*, v4i AS3*, imm offset, imm cpol)
    __builtin_amdgcn_global_load_async_to_lds_b128(
        (__attribute__((address_space(1))) v4i*)const_cast<float*>(g),
        (__attribute__((address_space(3))) v4i*)l,
        0, 0);
#else
    *(float4*)l = *(const float4*)g;   // synchronous fallback
#endif
}

__device__ __forceinline__ void wait_async_all() {
#ifdef HAVE_ASYNC_LDS
#if __has_builtin(__builtin_amdgcn_s_wait_asynccnt)
    __builtin_amdgcn_s_wait_asynccnt(0);
#else
    asm volatile("s_wait_asynccnt 0x0" ::: "memory");
#endif
#endif
}

// max3(a,b,acc): backend folds fmax(fmax(a,b),acc) into v_max3_f32
__device__ __forceinline__ float max3(float a, float b, float acc) {
    return fmaxf(fmaxf(a, b), acc);
}

__global__ __launch_bounds__(256)
void tropical_linear_kernel(const float* __restrict__ x,
                            const float* __restrict__ w,
                            const float* __restrict__ bias,
                            float* __restrict__ out,
                            int IN, int OUT, int obshift)
{
    __shared__ float ws[2][TO * KCP];
    __shared__ float xs[2][TN * KCP];

    const int tid     = threadIdx.x;              // 0..255
    const int o_local = tid & 15;                 // 0..15
    const int n_local = tid >> 4;                 // 0..15
    const int ob      = blockIdx.x & ((1 << obshift) - 1);
    const int nb      = blockIdx.x >> obshift;
    const int o_base  = ob * TO;
    const int n_base  = nb * TN;

    const int nchunks = IN / KC;

    // Stage one K-chunk: 32 rows x 16 float4 for each of w and x
    // = 1024 float4 total, 4 async b128 issues per thread, 512B/wave-instr.
    auto issue_chunk = [&](int kc, int buf) {
#pragma unroll
        for (int j = 0; j < 2; ++j) {
            const int idx = tid + 256 * j;        // 0..511
            const int row = idx >> 4;             // 0..31
            const int c4  = idx & 15;             // 0..15
            async_copy16(w + (size_t)(o_base + row) * IN + kc + c4 * 4,
                         &ws[buf][row * KCP + c4 * 4]);
        }
#pragma unroll
        for (int j = 0; j < 2; ++j) {
            const int idx = tid + 256 * j;
            const int row = idx >> 4;
            const int c4  = idx & 15;
            async_copy16(x + (size_t)(n_base + row) * IN + kc + c4 * 4,
                         &xs[buf][row * KCP + c4 * 4]);
        }
    };

    issue_chunk(0, 0);
    wait_async_all();
    __syncthreads();

    float a00 = -INFINITY, a01 = -INFINITY, a10 = -INFINITY, a11 = -INFINITY;

    for (int c = 0; c < nchunks; ++c) {
        const int buf = c & 1;
        if (c + 1 < nchunks)
            issue_chunk((c + 1) * KC, buf ^ 1);   // overlaps compute below

        const float* w0 = &ws[buf][o_local * KCP];          // row o_local
        const float* w1 = w0 + 16 * KCP;                    // row o_local+16
        const float* x0 = &xs[buf][n_local * KCP];          // row n_local
        const float* x1 = x0 + 16 * KCP;                    // row n_local+16
#pragma unroll
        for (int i = 0; i < KC; i += 4) {
            const float4 wv0 = *(const float4*)(w0 + i);
            const float4 wv1 = *(const float4*)(w1 + i);
            const float4 xv0 = *(const float4*)(x0 + i);
            const float4 xv1 = *(const float4*)(x1 + i);

            a00 = max3(wv0.x + xv0.x, wv0.y + xv0.y, a00);
            a00 = max3(wv0.z + xv0.z, wv0.w + xv0.w, a00);
            a01 = max3(wv1.x + xv0.x, wv1.y + xv0.y, a01);
            a01 = max3(wv1.z + xv0.z, wv1.w + xv0.w, a01);
            a10 = max3(wv0.x + xv1.x, wv0.y + xv1.y, a10);
            a10 = max3(wv0.z + xv1.z, wv0.w + xv1.w, a10);
            a11 = max3(wv1.x + xv1.x, wv1.y + xv1.y, a11);
            a11 = max3(wv1.z + xv1.z, wv1.w + xv1.w, a11);
        }

        wait_async_all();       // chunk c+1 copies fully landed in LDS
        __syncthreads();        // all waves done reading buf & done copying
    }

    const int o0 = o_base + o_local, o1 = o0 + 16;
    const int n0 = n_base + n_local, n1 = n0 + 16;
    const float b0 = bias[o0], b1 = bias[o1];
    out[(size_t)n0 * OUT + o0] = a00 + b0;
    out[(size_t)n0 * OUT + o1] = a01 + b1;
    out[(size_t)n1 * OUT + o0] = a10 + b0;
    out[(size_t)n1 * OUT + o1] = a11 + b1;
}

extern "C" void kernel_launch(void* const* d_in, const int* in_sizes, int n_in,
                              void* d_out, int out_size, void* d_ws, size_t ws_size,
                              hipStream_t stream) {
    const float* x = (const float*)d_in[0];   // [N, IN]
    const float* w = (const float*)d_in[1];   // [OUT, IN]
    const float* b = (const float*)d_in[2];   // [OUT]
    float* out = (float*)d_out;               // [N, OUT]

    const int OUT = in_sizes[2];              // 1024
    const int IN  = in_sizes[1] / OUT;        // 1024
    const int N   = in_sizes[0] / IN;         // 128

    const int oblocks = OUT / TO;             // 32 (power of 2)
    const int nblocks = N / TN;               // 4
    int obshift = 0;
    while ((1 << obshift) < oblocks) ++obshift;   // 5

    dim3 grid(oblocks * nblocks);             // 128 blocks
    dim3 block(256);                          // 8 wave32s

    tropical_linear_kernel<<<grid, block, 0, stream>>>(x, w, b, out, IN, OUT, obshift);
}